// LSSViewTransform_80564996539156
// MI455X (gfx1250) — compile-verified
//
#include <hip/hip_runtime.h>

typedef float v2f __attribute__((ext_vector_type(2)));
typedef float v4f __attribute__((ext_vector_type(4)));
typedef float v8f __attribute__((ext_vector_type(8)));

#define IN_C   256
#define CTX_C  128
#define H_IN   32
#define W_IN   88
#define HW_IN  (H_IN * W_IN)     // 2816
#define BATCH  4
#define BEVW   432               // maps H_IN
#define BEVH   496               // maps W_IN

// ---------------------------------------------------------------------------
// Kernel 1: ctx[b][c][hw] = (sum_k Wc[64+c][k] * feat[b][k][hw] + bc[64+c]) / 64
// fp32 WMMA 16x16x4. One block = one N-tile of 16 pixels; 8 waves cover the
// 128 context channels (wave w -> channels 16w..16w+15).
// ---------------------------------------------------------------------------
__global__ __launch_bounds__(256) void lss_gemm_ctx_kernel(
    const float* __restrict__ feat, const float* __restrict__ Wc,
    const float* __restrict__ bc, float* __restrict__ ctx)
{
  // B tile staged in LDS, pair-interleaved: element (k, n) at (k>>1)*32 + 2n + (k&1)
  __shared__ float Bs[IN_C * 16];

  const int tile = blockIdx.x;                 // 0..703
  const int b    = tile / (HW_IN / 16);        // tile / 176
  const int hw0  = (tile % (HW_IN / 16)) * 16;
  const int tid  = threadIdx.x;

  const float* fbase = feat + (size_t)b * IN_C * HW_IN + hw0;
#pragma unroll
  for (int r = 0; r < 16; ++r) {
    int e = r * 256 + tid;
    int k = e >> 4;
    int n = e & 15;
    Bs[(k >> 1) * 32 + n * 2 + (k & 1)] = fbase[(size_t)k * HW_IN + n];
  }
  __syncthreads();

  const int wave = tid >> 5;
  const int lane = tid & 31;
  const int m    = lane & 15;   // A-matrix row held by this lane
  const int kh   = lane >> 4;   // K-half select (0: K=0,1  1: K=2,3)

  // A: Wc rows 64 + wave*16 + m; lane reads K = 4*kk + 2*kh (+1), contiguous 8B
  const float* arow = Wc + (size_t)(64 + wave * 16 + m) * IN_C + 2 * kh;
  // B: lds address (2*kk + kh)*32 + 2n  ->  brow + kk*64
  const float* brow = Bs + kh * 32 + 2 * (lane & 15);

  v8f acc = (v8f){0.f, 0.f, 0.f, 0.f, 0.f, 0.f, 0.f, 0.f};

#pragma unroll 8
  for (int kk = 0; kk < 64; ++kk) {
    v2f a, bvec;
    a.x = arow[4 * kk + 0];
    a.y = arow[4 * kk + 1];
    const float* bp = brow + (size_t)kk * 64;
    bvec.x = bp[0];
    bvec.y = bp[1];
    // D = A(16x4) * B(4x16) + C   -> v_wmma_f32_16x16x4_f32
    acc = __builtin_amdgcn_wmma_f32_16x16x4_f32(
        /*neg_a=*/false, a, /*neg_b=*/false, bvec,
        /*c_mod=*/(short)0, acc, /*reuse_a=*/false, /*reuse_b=*/false);
  }

  // D layout: vgpr r, lane l -> (m = r + 8*(l>>4), n = l&15)
  float* obase = ctx + (size_t)b * CTX_C * HW_IN + hw0 + (lane & 15);
#pragma unroll
  for (int r = 0; r < 8; ++r) {
    int c  = wave * 16 + r + 8 * kh;                       // ctx channel
    float v = (acc[r] + bc[64 + c]) * (1.0f / 64.0f);      // mean(softmax)==1/64
    obase[(size_t)c * HW_IN] = v;
  }
}

// ---------------------------------------------------------------------------
// Kernel 2: bilinear resize (4,128,32,88) -> (4,128,432,496), half-pixel
// centers with edge clamp (== jax.image.resize bilinear upsample).
// Each thread produces 4 consecutive x outputs, 16B non-temporal store.
// ---------------------------------------------------------------------------
__global__ __launch_bounds__(256) void lss_resize_kernel(
    const float* __restrict__ ctx, float* __restrict__ out)
{
  const int g    = blockIdx.x * 256 + threadIdx.x;
  const int qx   = g % (BEVH / 4);           // 0..123
  const int t    = g / (BEVH / 4);
  const int oy   = t % BEVW;                 // 0..431
  const int bcid = t / BEVW;                 // 0..511  (b*128 + c)
  if (bcid >= BATCH * CTX_C) return;

  float sy = (oy + 0.5f) * ((float)H_IN / (float)BEVW) - 0.5f;
  sy = fminf(fmaxf(sy, 0.0f), (float)(H_IN - 1));
  int   y0 = (int)sy;
  int   y1 = min(y0 + 1, H_IN - 1);
  float wy = sy - (float)y0;

  const float* base = ctx + (size_t)bcid * HW_IN;
  const float* r0   = base + y0 * W_IN;
  const float* r1   = base + y1 * W_IN;

  float rr[4];
#pragma unroll
  for (int j = 0; j < 4; ++j) {
    int   ox = qx * 4 + j;
    float sx = (ox + 0.5f) * ((float)W_IN / (float)BEVH) - 0.5f;
    sx = fminf(fmaxf(sx, 0.0f), (float)(W_IN - 1));
    int   x0 = (int)sx;
    int   x1 = min(x0 + 1, W_IN - 1);
    float wx = sx - (float)x0;

    float v00 = r0[x0], v01 = r0[x1];
    float v10 = r1[x0], v11 = r1[x1];
    float top = v00 + wx * (v01 - v00);
    float bot = v10 + wx * (v11 - v10);
    rr[j] = top + wy * (bot - top);
  }

  v4f res = (v4f){rr[0], rr[1], rr[2], rr[3]};
  v4f* dst = (v4f*)(out + ((size_t)bcid * BEVW + oy) * BEVH + (size_t)qx * 4);
  __builtin_nontemporal_store(res, dst);   // stream 439MB out, skip cache
}

// ---------------------------------------------------------------------------
extern "C" void kernel_launch(void* const* d_in, const int* in_sizes, int n_in,
                              void* d_out, int out_size, void* d_ws, size_t ws_size,
                              hipStream_t stream) {
  const float* feat = (const float*)d_in[0];   // (4,256,32,88)
  const float* Wc   = (const float*)d_in[1];   // (192,256)
  const float* bc   = (const float*)d_in[2];   // (192,)
  float* out = (float*)d_out;                  // (4,128,432,496)
  float* ctx = (float*)d_ws;                   // scratch: 4*128*2816 floats (5.5 MB)

  // 704 N-tiles of 16 pixels; 256 threads = 8 waves covering 128 channels
  lss_gemm_ctx_kernel<<<704, 256, 0, stream>>>(feat, Wc, bc, ctx);

  // 512 * 432 * 124 threads / 256 = 107136 blocks (exact)
  const int total  = BATCH * CTX_C * BEVW * (BEVH / 4);
  lss_resize_kernel<<<total / 256, 256, 0, stream>>>(ctx, out);
}